// ETypeClusModel_33277406609670
// MI455X (gfx1250) — compile-verified
//
#include <hip/hip_runtime.h>
#include <hip/hip_bf16.h>
#include <math.h>

typedef __attribute__((ext_vector_type(2))) float v2f;
typedef __attribute__((ext_vector_type(8))) float v8f;

#define EPSF  1e-12f
#define TEMPF 0.1f
#define BATCH 128
#define SEQL  256
#define DIM   1024

// ---------------------------------------------------------------------------
// Kernel 1: stable compaction (argsort of 1-mask) + span max-pool + l2norm
// One block per batch row; 256 threads; each thread owns 4 of the 1024 cols.
// This is the bandwidth-bound stage (<=134MB @ 23.3TB/s ~ 6us floor), so the
// row reads are fully coalesced and the next indirect row is prefetched.
// ---------------------------------------------------------------------------
__global__ void feat_kernel(const float* __restrict__ enc,   // [B,L,D]
                            const int*   __restrict__ vmask, // [B,L]
                            const int*   __restrict__ pspan, // [B,2]
                            float*       __restrict__ X)     // [B,D]
{
    const int b   = blockIdx.x;
    const int tid = threadIdx.x;          // 256 threads

    __shared__ int   vp[SEQL];
    __shared__ int   s_count;
    __shared__ float red[256];
    __shared__ float s_norm;

    if (tid == 0) {                       // stable compaction of valid indices
        int c = 0;
        const int* mrow = vmask + (size_t)b * SEQL;
        for (int l = 0; l < SEQL; ++l)
            if (mrow[l] != 0) vp[c++] = l;
        s_count = c;
    }
    __syncthreads();

    const int a   = pspan[2 * b];
    const int e   = pspan[2 * b + 1];
    const int cnt = s_count;
    const float* base = enc + (size_t)b * SEQL * DIM;

    float m[4];
#pragma unroll
    for (int j = 0; j < 4; ++j) m[j] = -INFINITY;

    for (int i = a; i <= e; ++i) {
        if (i < cnt) {
            const float* row = base + (size_t)vp[i] * DIM;
            if (i + 1 <= e && i + 1 < cnt) {
                // hide the gather indirection: prefetch next row's cacheline
                __builtin_prefetch(base + (size_t)vp[i + 1] * DIM + tid, 0, 3);
            }
#pragma unroll
            for (int j = 0; j < 4; ++j)
                m[j] = fmaxf(m[j], row[tid + j * 256]);
        } else {
            // beyond the valid count the compacted row is zero
#pragma unroll
            for (int j = 0; j < 4; ++j) m[j] = fmaxf(m[j], 0.0f);
        }
    }

    // block reduction of sum of squares for the L2 norm
    float s = 0.0f;
#pragma unroll
    for (int j = 0; j < 4; ++j) s += m[j] * m[j];
    red[tid] = s;
    __syncthreads();
    for (int off = 128; off > 0; off >>= 1) {
        if (tid < off) red[tid] += red[tid + off];
        __syncthreads();
    }
    if (tid == 0) s_norm = fmaxf(sqrtf(red[0]), EPSF);
    __syncthreads();

    const float inv = 1.0f / s_norm;
#pragma unroll
    for (int j = 0; j < 4; ++j)
        X[(size_t)b * DIM + tid + j * 256] = m[j] * inv;
}

// ---------------------------------------------------------------------------
// Kernel 2: fp32 WMMA GEMM  Y[M,N] = act(X[M,K] @ W[K,N] + bias)
// V_WMMA_F32_16X16X4_F32, K step = 4 (all model K dims % 4 == 0).
// Mapping: block = one 16-wide N-tile, the 8 waves = the 8 M-tiles of M=128.
// -> all 8 waves issue IDENTICAL W addresses (WGP$ broadcast; W streamed from
//    L2 once per block) while each wave reads distinct X rows.
// N edge is handled with clamped addresses + multiplicative mask so EXEC is
// all-ones at every WMMA (ISA requirement).
// ---------------------------------------------------------------------------
__global__ void gemm_wmma_f32(const float* __restrict__ Xin, // [M,K]
                              const float* __restrict__ W,   // [K,N]
                              const float* __restrict__ bias,// [N]
                              float*       __restrict__ Y,   // [M,N]
                              int K, int N, int relu)
{
    const int mt   = threadIdx.x >> 5;               // wave id = M tile (0..7)
    const int nt   = blockIdx.x;                     // block   = N tile
    const int lane = threadIdx.x & 31;

    const int   half  = lane >> 4;                   // 0 or 1
    const int   r     = lane & 15;
    const int   mrow  = mt * 16 + r;                 // A row  (M = lane%16)
    const int   ncol  = nt * 16 + r;                 // B col  (N = lane%16)
    const int   ncl   = (ncol < N) ? ncol : (N - 1);
    const float bmask = (ncol < N) ? 1.0f : 0.0f;

    // A: lanes 0-15 hold K=k0,k0+1 ; lanes 16-31 hold K=k0+2,k0+3.
    // mrow*K + 2*half is even and K%4==0 -> 8B-aligned v2f (b64) loads.
    const float* xp = Xin + (size_t)mrow * K + 2 * half;
    // B rows kA and kA+1 for column ncl; advance by 4 rows per step.
    const float* bp = W + (size_t)(2 * half) * N + ncl;

    v8f acc = {};
    for (int k0 = 0; k0 < K; k0 += 4) {
        v2f av = *(const v2f*)xp;
        v2f bv;
        bv.x = bp[0] * bmask;
        bv.y = bp[(size_t)N] * bmask;
        acc = __builtin_amdgcn_wmma_f32_16x16x4_f32(
                  /*neg_a=*/false, av, /*neg_b=*/false, bv,
                  /*c_mod=*/(short)0, acc, /*reuse_a=*/false, /*reuse_b=*/false);
        xp += 4;
        bp += (size_t)4 * N;
    }

    // C/D layout: VGPR r2 holds M = r2 + 8*half, N = lane%16
    if (ncol < N) {
        const float bb = bias[ncol];
#pragma unroll
        for (int r2 = 0; r2 < 8; ++r2) {
            const int row = mt * 16 + 8 * half + r2;
            float v = acc[r2] + bb;
            if (relu) v = fmaxf(v, 0.0f);
            Y[(size_t)row * N + ncol] = v;
        }
    }
}

// ---------------------------------------------------------------------------
// Kernel 3: p = softmax( l2norm(z) @ l2norm(topic_emb)^T / TEMP )
// One block per batch row; tiny (100-dim dot with 50 topics).
// ---------------------------------------------------------------------------
__global__ void topic_kernel(const float* __restrict__ Z,  // [B,Dz]
                             const float* __restrict__ T,  // [Kt,Dz]
                             float*       __restrict__ P,  // [B,Kt]
                             int Dz, int Kt)
{
    const int b   = blockIdx.x;
    const int tid = threadIdx.x;          // 64 threads

    __shared__ float zsh[128];
    __shared__ float logits[64];
    __shared__ float znorm;

    for (int i = tid; i < Dz; i += blockDim.x)
        zsh[i] = Z[(size_t)b * Dz + i];
    __syncthreads();

    if (tid == 0) {
        float s = 0.0f;
        for (int i = 0; i < Dz; ++i) s += zsh[i] * zsh[i];
        znorm = fmaxf(sqrtf(s), EPSF);
    }
    __syncthreads();

    for (int j = tid; j < Kt; j += blockDim.x) {
        const float* trow = T + (size_t)j * Dz;
        float tn = 0.0f, dot = 0.0f;
        for (int i = 0; i < Dz; ++i) {
            tn  += trow[i] * trow[i];
            dot += zsh[i] * trow[i];
        }
        tn = fmaxf(sqrtf(tn), EPSF);
        logits[j] = dot / (znorm * tn) / TEMPF;
    }
    __syncthreads();

    if (tid == 0) {
        float mx = -INFINITY;
        for (int j = 0; j < Kt; ++j) mx = fmaxf(mx, logits[j]);
        float s = 0.0f;
        for (int j = 0; j < Kt; ++j) { float ev = expf(logits[j] - mx); logits[j] = ev; s += ev; }
        const float inv = 1.0f / s;
        for (int j = 0; j < Kt; ++j) P[(size_t)b * Kt + j] = logits[j] * inv;
    }
}

// ---------------------------------------------------------------------------
// Kernel 4: per-row cos(b) = <l2norm(h_b), x_b> ; deterministic block reduce
// ---------------------------------------------------------------------------
__global__ void cos_kernel(const float* __restrict__ H,  // [B,D] pre-norm x_bar
                           const float* __restrict__ X,  // [B,D]
                           float*       __restrict__ cosv)
{
    const int b   = blockIdx.x;
    const int tid = threadIdx.x;          // 256 threads
    __shared__ float r1[256], r2[256];

    float shh = 0.0f, shx = 0.0f;
    for (int d = tid; d < DIM; d += 256) {
        const float h = H[(size_t)b * DIM + d];
        const float x = X[(size_t)b * DIM + d];
        shh += h * h;
        shx += h * x;
    }
    r1[tid] = shh; r2[tid] = shx;
    __syncthreads();
    for (int off = 128; off > 0; off >>= 1) {
        if (tid < off) { r1[tid] += r1[tid + off]; r2[tid] += r2[tid + off]; }
        __syncthreads();
    }
    if (tid == 0) cosv[b] = r2[0] / fmaxf(sqrtf(r1[0]), EPSF);
}

__global__ void loss_kernel(const float* __restrict__ cosv, float* __restrict__ out)
{
    __shared__ float red[128];
    const int tid = threadIdx.x;          // 128 threads
    red[tid] = 1.0f - cosv[tid];
    __syncthreads();
    for (int off = 64; off > 0; off >>= 1) {
        if (tid < off) red[tid] += red[tid + off];
        __syncthreads();
    }
    if (tid == 0) out[0] = red[0] / (float)BATCH;
}

// ---------------------------------------------------------------------------
extern "C" void kernel_launch(void* const* d_in, const int* in_sizes, int n_in,
                              void* d_out, int out_size, void* d_ws, size_t ws_size,
                              hipStream_t stream) {
    // setup_inputs() order, lists flattened element-wise:
    const float* enc   = (const float*)d_in[0];    // [128,256,1024]
    const int*   vmask = (const int*)  d_in[1];    // [128,256]
    const int*   pspan = (const int*)  d_in[2];    // [128,2]
    // d_in[3] = mask_span (unused by the reference computation)
    const float* eW[4] = { (const float*)d_in[4],  (const float*)d_in[5],
                           (const float*)d_in[6],  (const float*)d_in[7]  };
    const float* eb[4] = { (const float*)d_in[8],  (const float*)d_in[9],
                           (const float*)d_in[10], (const float*)d_in[11] };
    const float* dW[4] = { (const float*)d_in[12], (const float*)d_in[13],
                           (const float*)d_in[14], (const float*)d_in[15] };
    const float* db[4] = { (const float*)d_in[16], (const float*)d_in[17],
                           (const float*)d_in[18], (const float*)d_in[19] };
    const float* temb  = (const float*)d_in[20];   // [50,100]

    float* out  = (float*)d_out;                   // [0]=loss, [1..]=p (128x50)
    float* X    = (float*)d_ws;                    // 128*1024
    float* T0   = X  + BATCH * DIM;                // 128*1024
    float* T1   = T0 + BATCH * DIM;                // 128*1024
    float* cosv = T1 + BATCH * DIM;                // 128

    feat_kernel<<<BATCH, 256, 0, stream>>>(enc, vmask, pspan, X);

    auto gemm = [&](const float* A, const float* Wm, const float* bv, float* Yv,
                    int K, int N, int relu) {
        const int Nt = (N + 15) / 16;              // blocks = N tiles
        gemm_wmma_f32<<<Nt, 256, 0, stream>>>(A, Wm, bv, Yv, K, N, relu);
    };

    // encoder: relu on all but last
    gemm(X,  eW[0], eb[0], T0, 1024, 500, 1);
    gemm(T0, eW[1], eb[1], T1, 500,  500, 1);
    gemm(T1, eW[2], eb[2], T0, 500, 1000, 1);
    gemm(T0, eW[3], eb[3], T1, 1000, 100, 0);      // z in T1 (128x100)

    topic_kernel<<<BATCH, 64, 0, stream>>>(T1, temb, out + 1, 100, 50);

    // decoder: relu on all but last
    gemm(T1, dW[0], db[0], T0, 100, 1000, 1);
    gemm(T0, dW[1], db[1], T1, 1000, 500, 1);
    gemm(T1, dW[2], db[2], T0, 500,  500, 1);
    gemm(T0, dW[3], db[3], T1, 500, 1024, 0);      // h (pre-norm x_bar) in T1

    cos_kernel<<<BATCH, 256, 0, stream>>>(T1, X, cosv);
    loss_kernel<<<1, 128, 0, stream>>>(cosv, out);
}